// NTMHead_5291399709228
// MI455X (gfx1250) — compile-verified
//
#include <hip/hip_runtime.h>
#include <math.h>

#define NSLOTS  131072
#define DMEM    64
#define BATCH   256
#define DIN     256
#define DOUTC   70
#define EPSV    1e-8f

#define SPAN    512                 // slots per block in the big passes
#define SBLKS   (NSLOTS / SPAN)     // 256 slot-blocks
#define BTILES  (BATCH / 16)        // 16 batch tiles of 16 rows
#define CHUNK   64                  // slots staged in LDS per iteration
#define NWAVES  4
#define THREADS 128                 // 4 waves (wave32)
#define LSTRIDE 66                  // padded LDS row stride (bank-conflict free)

typedef float v2f __attribute__((ext_vector_type(2)));
typedef float v8f __attribute__((ext_vector_type(8)));
typedef int   v4i __attribute__((ext_vector_type(4)));
// pointer-to-AS-qualified-v4i types for the async LDS builtin
typedef __attribute__((address_space(1))) v4i* gptr_v4i;
typedef __attribute__((address_space(3))) v4i* lptr_v4i;

// ---- workspace layout (float offsets) ----
#define OFF_KEY   0                         // BATCH*DMEM   = 16384
#define OFF_BETA  16384                     // BATCH
#define OFF_GATE  16640                     // BATCH
#define OFF_KN    16896                     // BATCH
#define OFF_CK    17152                     // BATCH*3      = 768
#define OFF_MN    17920                     // NSLOTS       = 131072
#define OFF_PM    148992                    // BATCH*SBLKS  = 65536
#define OFF_PS    214528                    // BATCH*SBLKS  = 65536
#define OFF_RMAX  280064                    // BATCH
#define OFF_RSUM  280320                    // BATCH
// total ~280576 floats = ~1.1 MB

#if __has_builtin(__builtin_amdgcn_global_load_async_to_lds_b128)
#define HAS_ASYNC_LDS 1
#else
#define HAS_ASYNC_LDS 0
#endif

__device__ inline void wait_asynccnt0() {
#if __has_builtin(__builtin_amdgcn_s_wait_asynccnt)
  __builtin_amdgcn_s_wait_asynccnt(0);
#else
  asm volatile("s_wait_asynccnt 0" ::: "memory");
#endif
}

// Stage CHUNK(64) memory rows (64 f32 each) into padded LDS tile.
// Each 16B segment stays inside one row, so the LSTRIDE padding is preserved.
__device__ inline void stage_mem_chunk(const float* __restrict__ mem, int cbase,
                                       float* __restrict__ memL, int tid) {
#if HAS_ASYNC_LDS
  #pragma unroll
  for (int i = 0; i < (CHUNK * DMEM) / (THREADS * 4); ++i) {   // 8 issues
    const int idx = tid * 4 + i * THREADS * 4;
    const int r = idx >> 6, c = idx & 63;
    gptr_v4i g = (gptr_v4i)(mem + (size_t)(cbase + r) * DMEM + c);
    lptr_v4i l = (lptr_v4i)(&memL[r * LSTRIDE + c]);
    __builtin_amdgcn_global_load_async_to_lds_b128(g, l, 0, 0);
  }
  wait_asynccnt0();
#else
  for (int i = tid * 4; i < CHUNK * DMEM; i += THREADS * 4) {
    const int r = i >> 6, c = i & 63;
    const float4 v = *(const float4*)(mem + (size_t)(cbase + r) * DMEM + c);
    memL[r * LSTRIDE + c + 0] = v.x; memL[r * LSTRIDE + c + 1] = v.y;
    memL[r * LSTRIDE + c + 2] = v.z; memL[r * LSTRIDE + c + 3] = v.w;
  }
#endif
}

// ------------------------------------------------------------------
// Kernel 1: combined = x @ W + b ; derive key, beta, gate, conv-softmax, |key|
// ------------------------------------------------------------------
__global__ __launch_bounds__(256)
void k_setup(const float* __restrict__ x, const float* __restrict__ W,
             const float* __restrict__ bvec, float* __restrict__ ws) {
  __shared__ float xr[DIN];
  __shared__ float comb[DOUTC];
  const int b = blockIdx.x, t = threadIdx.x;
  xr[t] = x[b * DIN + t];
  __syncthreads();
  if (t < 69) {                       // 64 key + beta + gate + 3 conv taps
    float acc = bvec[t];
    for (int k = 0; k < DIN; ++k) acc += xr[k] * W[k * DOUTC + t];
    comb[t] = acc;
    if (t < DMEM) ws[OFF_KEY + b * DMEM + t] = acc;
  }
  __syncthreads();
  if (t == 0) {
    float s = 0.f;
    for (int k = 0; k < DMEM; ++k) s += comb[k] * comb[k];
    ws[OFF_KN + b]   = sqrtf(s);
    ws[OFF_BETA + b] = comb[64];
    ws[OFF_GATE + b] = 1.f / (1.f + __expf(-comb[65]));
    const float m  = fmaxf(comb[66], fmaxf(comb[67], comb[68]));
    const float e0 = __expf(comb[66] - m), e1 = __expf(comb[67] - m), e2 = __expf(comb[68] - m);
    const float inv = 1.f / (e0 + e1 + e2);
    ws[OFF_CK + b * 3 + 0] = e0 * inv;
    ws[OFF_CK + b * 3 + 1] = e1 * inv;
    ws[OFF_CK + b * 3 + 2] = e2 * inv;
  }
}

// ------------------------------------------------------------------
// Kernel 2: mem_norm[n] = ||memory[n,:]||
// ------------------------------------------------------------------
__global__ __launch_bounds__(256)
void k_memnorm(const float* __restrict__ mem, float* __restrict__ ws) {
  const int n = blockIdx.x * blockDim.x + threadIdx.x;
  const float* r = mem + (size_t)n * DMEM;
  float s = 0.f;
  for (int k = 0; k < DMEM; k += 4) {
    const float4 v = *(const float4*)(r + k);
    s += v.x * v.x + v.y * v.y + v.z * v.z + v.w * v.w;
  }
  ws[OFF_MN + n] = sqrtf(s);
}

// 16-lane butterfly reductions (rows live across 16-lane halves)
__device__ inline float redmax16(float v) {
  v = fmaxf(v, __shfl_xor(v, 1, 32));
  v = fmaxf(v, __shfl_xor(v, 2, 32));
  v = fmaxf(v, __shfl_xor(v, 4, 32));
  v = fmaxf(v, __shfl_xor(v, 8, 32));
  return v;
}
__device__ inline float redsum16(float v) {
  v += __shfl_xor(v, 1, 32);
  v += __shfl_xor(v, 2, 32);
  v += __shfl_xor(v, 4, 32);
  v += __shfl_xor(v, 8, 32);
  return v;
}

// ------------------------------------------------------------------
// Kernel 3: WMMA dots -> scores -> per-(row, slot-block) online (max, sumexp)
// ------------------------------------------------------------------
__global__ __launch_bounds__(THREADS)
void k_scores(const float* __restrict__ mem, float* __restrict__ ws) {
  __shared__ float keyL[16 * LSTRIDE];
  __shared__ float memL[CHUNK * LSTRIDE];
  __shared__ float betaL[16], knL[16];
  __shared__ float wms[NWAVES][16][2];

  const int tid  = threadIdx.x;
  const int wave = tid >> 5;
  const int lane = tid & 31;
  const int hg   = lane >> 4;          // half-group: selects K pair / row+8
  const int ln   = lane & 15;
  const int bt   = blockIdx.y;
  const int base = blockIdx.x * SPAN;

  for (int i = tid; i < 16 * DMEM; i += THREADS) {
    const int r = i >> 6, c = i & 63;
    keyL[r * LSTRIDE + c] = ws[OFF_KEY + (bt * 16 + r) * DMEM + c];
  }
  if (tid < 16) { betaL[tid] = ws[OFF_BETA + bt * 16 + tid];
                  knL[tid]   = ws[OFF_KN   + bt * 16 + tid]; }

  float mrun[8], srun[8];
  #pragma unroll
  for (int i = 0; i < 8; ++i) { mrun[i] = -3.0e38f; srun[i] = 0.f; }

  for (int ch = 0; ch < SPAN / CHUNK; ++ch) {
    const int cbase = base + ch * CHUNK;
    __syncthreads();
    stage_mem_chunk(mem, cbase, memL, tid);
    __syncthreads();

    const int trow = wave * 16;                        // this wave's slot tile
    const float mnv = ws[OFF_MN + cbase + trow + ln];  // mem_norm for column ln
    v8f acc = {0.f, 0.f, 0.f, 0.f, 0.f, 0.f, 0.f, 0.f};
    #pragma unroll
    for (int k0 = 0; k0 < DMEM; k0 += 4) {
      const int kk = k0 + hg * 2;
      const v2f a = *(const v2f*)&keyL[ln * LSTRIDE + kk];
      const v2f b = *(const v2f*)&memL[(trow + ln) * LSTRIDE + kk];
      acc = __builtin_amdgcn_wmma_f32_16x16x4_f32(false, a, false, b,
                                                  (short)0, acc, false, false);
    }
    #pragma unroll
    for (int i = 0; i < 8; ++i) {
      const int row = i + 8 * hg;
      // 1-ulp reciprocal instead of IEEE division: scores only feed softmax
      const float invd = __builtin_amdgcn_rcpf(knL[row] * mnv + EPSV);
      const float sc = betaL[row] * acc[i] * invd;
      const float mt = redmax16(sc);
      const float st = redsum16(__expf(sc - mt));
      const float nm = fmaxf(mrun[i], mt);
      srun[i] = srun[i] * __expf(mrun[i] - nm) + st * __expf(mt - nm);
      mrun[i] = nm;
    }
  }
  __syncthreads();
  if (ln == 0) {
    #pragma unroll
    for (int i = 0; i < 8; ++i) {
      const int row = i + 8 * hg;
      wms[wave][row][0] = mrun[i]; wms[wave][row][1] = srun[i];
    }
  }
  __syncthreads();
  if (tid < 16) {
    float M = -3.0e38f, S = 0.f;
    for (int w = 0; w < NWAVES; ++w) {
      const float m = wms[w][tid][0], s = wms[w][tid][1];
      const float nm = fmaxf(M, m);
      S = S * __expf(M - nm) + s * __expf(m - nm);
      M = nm;
    }
    ws[OFF_PM + (bt * 16 + tid) * SBLKS + blockIdx.x] = M;
    ws[OFF_PS + (bt * 16 + tid) * SBLKS + blockIdx.x] = S;
  }
}

// ------------------------------------------------------------------
// Kernel 4: combine slot-block partials -> per-row (max, sumexp)
// ------------------------------------------------------------------
__global__ __launch_bounds__(256)
void k_reduce(float* __restrict__ ws) {
  const int r = threadIdx.x;
  float M = -3.0e38f, S = 0.f;
  for (int sb = 0; sb < SBLKS; ++sb) {
    const float m = ws[OFF_PM + r * SBLKS + sb];
    const float s = ws[OFF_PS + r * SBLKS + sb];
    const float nm = fmaxf(M, m);
    S = S * __expf(M - nm) + s * __expf(m - nm);
    M = nm;
  }
  ws[OFF_RMAX + r] = M;
  ws[OFF_RSUM + r] = S;
}

// ------------------------------------------------------------------
// Kernel 5: recompute dots (WMMA, L2-resident memory), w_c -> w_g in LDS
//           (+ 1-slot circular halo), fused 3-tap circular conv, NT store.
// ------------------------------------------------------------------
__global__ __launch_bounds__(THREADS)
void k_final(const float* __restrict__ mem, const float* __restrict__ prev,
             float* __restrict__ out, const float* __restrict__ ws) {
  __shared__ float keyL[16 * LSTRIDE];
  __shared__ float memL[CHUNK * LSTRIDE];
  __shared__ float betaL[16], knL[16], gateL[16], rmL[16], rsL[16];
  __shared__ float ckL[16][3];
  __shared__ float wg[16][SPAN + 2];

  const int tid  = threadIdx.x;
  const int wave = tid >> 5;
  const int lane = tid & 31;
  const int hg   = lane >> 4;
  const int ln   = lane & 15;
  const int bt   = blockIdx.y;
  const int base = blockIdx.x * SPAN;

  for (int i = tid; i < 16 * DMEM; i += THREADS) {
    const int r = i >> 6, c = i & 63;
    keyL[r * LSTRIDE + c] = ws[OFF_KEY + (bt * 16 + r) * DMEM + c];
  }
  if (tid < 16) {
    betaL[tid] = ws[OFF_BETA + bt * 16 + tid];
    knL[tid]   = ws[OFF_KN   + bt * 16 + tid];
    gateL[tid] = ws[OFF_GATE + bt * 16 + tid];
    rmL[tid]   = ws[OFF_RMAX + bt * 16 + tid];
    rsL[tid]   = 1.f / ws[OFF_RSUM + bt * 16 + tid];
    ckL[tid][0] = ws[OFF_CK + (bt * 16 + tid) * 3 + 0];
    ckL[tid][1] = ws[OFF_CK + (bt * 16 + tid) * 3 + 1];
    ckL[tid][2] = ws[OFF_CK + (bt * 16 + tid) * 3 + 2];
  }

  for (int ch = 0; ch < SPAN / CHUNK; ++ch) {
    const int cbase = base + ch * CHUNK;
    __syncthreads();
    stage_mem_chunk(mem, cbase, memL, tid);
    __syncthreads();

    const int trow = wave * 16;
    const float mnv = ws[OFF_MN + cbase + trow + ln];
    v8f acc = {0.f, 0.f, 0.f, 0.f, 0.f, 0.f, 0.f, 0.f};
    #pragma unroll
    for (int k0 = 0; k0 < DMEM; k0 += 4) {
      const int kk = k0 + hg * 2;
      const v2f a = *(const v2f*)&keyL[ln * LSTRIDE + kk];
      const v2f b = *(const v2f*)&memL[(trow + ln) * LSTRIDE + kk];
      acc = __builtin_amdgcn_wmma_f32_16x16x4_f32(false, a, false, b,
                                                  (short)0, acc, false, false);
    }
    #pragma unroll
    for (int i = 0; i < 8; ++i) {
      const int row  = i + 8 * hg;
      const int slot = cbase + trow + ln;
      const float invd = __builtin_amdgcn_rcpf(knL[row] * mnv + EPSV);
      const float sc = betaL[row] * acc[i] * invd;
      const float wc = __expf(sc - rmL[row]) * rsL[row];
      const float g  = gateL[row];
      // streaming read: keep memory matrix resident in L2, don't cache prev
      const float pv =
          __builtin_nontemporal_load(&prev[(size_t)(bt * 16 + row) * NSLOTS + slot]);
      wg[row][slot - base + 1] = g * wc + (1.f - g) * pv;
    }
  }
  __syncthreads();

  // circular halo: slot base-1 (left) and base+SPAN (right), mod N
  if (tid < 32) {
    const int r = tid & 15, side = tid >> 4;
    const int slot = side ? ((base + SPAN) & (NSLOTS - 1))
                          : ((base + NSLOTS - 1) & (NSLOTS - 1));
    const float* mr = mem + (size_t)slot * DMEM;
    float dot = 0.f;
    for (int k = 0; k < DMEM; ++k) dot += keyL[r * LSTRIDE + k] * mr[k];
    const float sc = betaL[r] * dot *
                     __builtin_amdgcn_rcpf(knL[r] * ws[OFF_MN + slot] + EPSV);
    const float wc = __expf(sc - rmL[r]) * rsL[r];
    const float pv = prev[(size_t)(bt * 16 + r) * NSLOTS + slot];
    wg[r][side ? (SPAN + 1) : 0] = gateL[r] * wc + (1.f - gateL[r]) * pv;
  }
  __syncthreads();

  // fused 3-tap circular conv: out[n] = ck0*wg[n-1] + ck1*wg[n] + ck2*wg[n+1]
  for (int idx = tid; idx < 16 * SPAN; idx += THREADS) {
    const int r = idx >> 9;          // SPAN == 512
    const int j = idx & (SPAN - 1);
    const float v = ckL[r][0] * wg[r][j] + ckL[r][1] * wg[r][j + 1] +
                    ckL[r][2] * wg[r][j + 2];
    __builtin_nontemporal_store(v, &out[(size_t)(bt * 16 + r) * NSLOTS + base + j]);
  }
}

// ------------------------------------------------------------------
extern "C" void kernel_launch(void* const* d_in, const int* in_sizes, int n_in,
                              void* d_out, int out_size, void* d_ws, size_t ws_size,
                              hipStream_t stream) {
  (void)in_sizes; (void)n_in; (void)out_size; (void)ws_size;
  const float* x    = (const float*)d_in[0];   // (256,256)
  const float* mem  = (const float*)d_in[1];   // (131072,64)
  const float* prev = (const float*)d_in[2];   // (256,131072)
  const float* W    = (const float*)d_in[3];   // (256,70)
  const float* bvec = (const float*)d_in[4];   // (70,)
  float* out = (float*)d_out;                  // (256,131072)
  float* ws  = (float*)d_ws;

  k_setup  <<<BATCH, 256, 0, stream>>>(x, W, bvec, ws);
  k_memnorm<<<NSLOTS / 256, 256, 0, stream>>>(mem, ws);
  k_scores <<<dim3(SBLKS, BTILES), THREADS, 0, stream>>>(mem, ws);
  k_reduce <<<1, 256, 0, stream>>>(ws);
  k_final  <<<dim3(SBLKS, BTILES), THREADS, 0, stream>>>(mem, prev, out, ws);
}